// EagleLlama3_1Attention_61615600828817
// MI455X (gfx1250) — compile-verified
//
#include <hip/hip_runtime.h>
#include <hip/hip_bf16.h>

typedef __attribute__((ext_vector_type(2))) float v2f;
typedef __attribute__((ext_vector_type(8))) float v8f;
typedef unsigned int u32x4 __attribute__((ext_vector_type(4)));
typedef int i32x4 __attribute__((ext_vector_type(4)));
typedef int i32x8 __attribute__((ext_vector_type(8)));

#define B_     16
#define S_     4
#define H_     4096
#define NH     32
#define NKV    8
#define HD     128
#define PAST   4096
#define TT     (PAST + S_)            // 4100
#define Q_SIZE (NH * HD)              // 4096
#define KV_SIZE (NKV * HD)            // 1024
#define QKV_N  (Q_SIZE + 2 * KV_SIZE) // 6144
#define NWAVE  2
#define NT_TILES ((TT + 15) / 16)     // 257
#define KROW   129                    // TDM pad: 1 dword after every 128 -> stride 129
#define KTILE_F (16 * KROW)           // floats per staged 16x128 tile

// ---------------------------------------------------------------------------
// Stream copy of past KV cache into the head of new_k / new_v.
// ---------------------------------------------------------------------------
__global__ void copy_past_kernel(const float4* __restrict__ pk,
                                 const float4* __restrict__ pv,
                                 float4* __restrict__ nk,
                                 float4* __restrict__ nv) {
    const size_t per_b = (size_t)PAST * NKV * HD / 4;
    const size_t dst_b = (size_t)TT * NKV * HD / 4;
    const size_t total = (size_t)B_ * per_b;
    size_t stride = (size_t)gridDim.x * blockDim.x;
    for (size_t i = (size_t)blockIdx.x * blockDim.x + threadIdx.x; i < total; i += stride) {
        size_t b = i / per_b, r = i - b * per_b;
        nk[b * dst_b + r] = pk[i];
        nv[b * dst_b + r] = pv[i];
    }
}

// ---------------------------------------------------------------------------
// C(M x N) = A(M x K) @ W(K x N), fp32, one wave per 16x16 tile,
// V_WMMA_F32_16X16X4_F32 over K.
// ---------------------------------------------------------------------------
__global__ __launch_bounds__(32) void gemm_f32_wmma(const float* __restrict__ A,
                                                    const float* __restrict__ W,
                                                    float* __restrict__ C,
                                                    int N, int K) {
    int nt = blockIdx.x, mt = blockIdx.y;
    int lane = threadIdx.x;
    int mn  = lane & 15;
    int kb  = (lane >> 4) << 1;   // 0 or 2

    const float* arow = A + (size_t)(mt * 16 + mn) * K;
    const float* wcol = W + nt * 16 + mn;

    v8f c = {};
    for (int k0 = 0; k0 < K; k0 += 64) {
        __builtin_prefetch(arow + k0 + 64, 0, 0);   // global_prefetch for next A chunk
#pragma unroll
        for (int kk = 0; kk < 64; kk += 4) {
            int kq = k0 + kk;
            v2f a = *(const v2f*)(arow + kq + kb);
            v2f b;
            b.x = wcol[(size_t)(kq + kb) * N];
            b.y = wcol[(size_t)(kq + kb + 1) * N];
            c = __builtin_amdgcn_wmma_f32_16x16x4_f32(false, a, false, b,
                                                      (short)0, c, false, false);
        }
    }
    int grp = lane >> 4;
    float* crow = C + nt * 16 + mn;
#pragma unroll
    for (int r = 0; r < 8; ++r) {
        int m = r + grp * 8;
        crow[(size_t)(mt * 16 + m) * N] = c[r];
    }
}

// ---------------------------------------------------------------------------
// RoPE on q (-> qbuf [b][g][row=s*4+r][128]) and k (-> new_k tail),
// plus copy of v into the new_v tail.
// ---------------------------------------------------------------------------
__global__ void rope_scatter_kernel(const int* __restrict__ positions,
                                    const float* __restrict__ qkv,
                                    float* __restrict__ qbuf,
                                    float* __restrict__ newk,
                                    float* __restrict__ newv) {
    const int QN = B_ * S_ * NH * (HD / 2);
    const int KN = B_ * S_ * NKV * (HD / 2);
    const int VN = B_ * S_ * NKV * HD;
    const float LOG_THETA = 13.122363377404328f; // log(500000)

    int idx = blockIdx.x * blockDim.x + threadIdx.x;
    if (idx < QN) {
        int i = idx & 63;
        int h = (idx >> 6) & 31;
        int s = (idx >> 11) & 3;
        int b = idx >> 13;
        float posf = (float)positions[b * S_ + s];
        float freq = posf * __expf((-2.0f * i / 128.0f) * LOG_THETA);
        float sn, cs; __sincosf(freq, &sn, &cs);
        const float* src = qkv + ((size_t)(b * S_ + s)) * QKV_N + h * HD;
        float x1 = src[i], x2 = src[i + 64];
        int g = h >> 2, r = h & 3, row = s * 4 + r;
        float* dst = qbuf + ((size_t)(b * NKV + g) * 16 + row) * HD;
        dst[i]      = x1 * cs - x2 * sn;
        dst[i + 64] = x2 * cs + x1 * sn;
    } else if (idx < QN + KN) {
        int j = idx - QN;
        int i = j & 63;
        int h = (j >> 6) & 7;
        int s = (j >> 9) & 3;
        int b = j >> 11;
        float posf = (float)positions[b * S_ + s];
        float freq = posf * __expf((-2.0f * i / 128.0f) * LOG_THETA);
        float sn, cs; __sincosf(freq, &sn, &cs);
        const float* src = qkv + ((size_t)(b * S_ + s)) * QKV_N + Q_SIZE + h * HD;
        float x1 = src[i], x2 = src[i + 64];
        float* dst = newk + (((size_t)b * TT + PAST + s) * NKV + h) * HD;
        dst[i]      = x1 * cs - x2 * sn;
        dst[i + 64] = x2 * cs + x1 * sn;
    } else if (idx < QN + KN + VN) {
        int j = idx - QN - KN;
        int i = j & 127;
        int h = (j >> 7) & 7;
        int s = (j >> 10) & 3;
        int b = j >> 12;
        newv[(((size_t)b * TT + PAST + s) * NKV + h) * HD + i] =
            qkv[((size_t)(b * S_ + s)) * QKV_N + Q_SIZE + KV_SIZE + h * HD + i];
    }
}

// ---------------------------------------------------------------------------
// TDM tile load: 16 rows x 128 f32, global row stride 1024 elements,
// LDS padded to 129-float rows (D# pad_enable, 1 dword per 128).
// D# packing per CDNA5 ISA 8.3/8.4.  rows = tensor_dim1 so OOB rows read 0.
// ---------------------------------------------------------------------------
#if __has_builtin(__builtin_amdgcn_tensor_load_to_lds)
#define HAVE_TDM 1
#else
#define HAVE_TDM 0
#endif

__device__ __forceinline__ void tdm_load_16x128(float* ldsptr, unsigned lds_byte,
                                                const float* gptr, unsigned rows) {
#if HAVE_TDM
    unsigned long long ga = (unsigned long long)(size_t)gptr;
    unsigned ga_lo = __builtin_amdgcn_readfirstlane((unsigned)ga);
    unsigned ga_hi = __builtin_amdgcn_readfirstlane((unsigned)(ga >> 32));
    unsigned lb    = __builtin_amdgcn_readfirstlane(lds_byte);
    unsigned rv    = __builtin_amdgcn_readfirstlane(rows);

    // Group 0: count=1 | lds_addr | global_addr(57b) | type=2
    u32x4 g0 = { 1u, lb, ga_lo, (ga_hi & 0x01ffffffu) | (2u << 30) };
    // Group 1: data_size=4B, pad_enable, pad_interval=128dw, pad_amount=1dw,
    // tensor_dim0=128, tensor_dim1=rows, tile_dim0=128, tile_dim1=16,
    // tensor_dim0_stride=1024 elements.
    i32x8 g1 = { (int)((2u << 16) | (1u << 20) | (6u << 22)),
                 (int)(128u << 16),          // tensor_dim0[15:0] @ bits 63:48
                 (int)((rv & 0xffffu) << 16),// dim0[31:16]=0 | tensor_dim1[15:0]
                 (int)(128u << 16),          // dim1[31:16]=0 | tile_dim0=128
                 16,                          // tile_dim1=16, tile_dim2=0
                 1024,                        // tensor_dim0_stride[31:0]
                 0, 0 };
    i32x4 z4 = { 0, 0, 0, 0 };
#if defined(__clang_major__) && (__clang_major__ >= 23)
    i32x8 z8 = { 0, 0, 0, 0, 0, 0, 0, 0 };
    __builtin_amdgcn_tensor_load_to_lds(g0, g1, z4, z4, z8, 0);
#else
    __builtin_amdgcn_tensor_load_to_lds(g0, g1, z4, z4, 0);
#endif
    __builtin_amdgcn_s_wait_tensorcnt(0);
    asm volatile("" ::: "memory");   // keep LDS reads below the tensorcnt wait
#else
    int lane = threadIdx.x & 31;
    for (int i = lane; i < 16 * 128; i += 32) {
        int r = i >> 7, c = i & 127;
        ldsptr[r * KROW + c] = (r < (int)rows) ? gptr[(size_t)r * 1024 + c] : 0.0f;
    }
#endif
}

// ---------------------------------------------------------------------------
// Flash attention, one workgroup per (b, kv_head).  Q block = 16 rows
// (S*reps).  2 waves strip-mine key tiles; K/V tiles staged to LDS by TDM;
// online softmax; wave partials combined via LDS.
// ---------------------------------------------------------------------------
__global__ __launch_bounds__(NWAVE * 32) void attn_kernel(
    const float* __restrict__ qbuf,   // [B][NKV][16][HD]
    const float* __restrict__ kc,     // new_k [B][TT][NKV][HD]
    const float* __restrict__ vc,     // new_v
    float* __restrict__ attn)         // [B][S][NH][HD]
{
    __shared__ float lds_k[NWAVE][KTILE_F + 16];
    __shared__ float lds_v[NWAVE][KTILE_F + 16];
    __shared__ float lds_p[NWAVE][16][17];
    __shared__ float lds_acc[NWAVE][16][HD];
    __shared__ float lds_m[NWAVE][16];
    __shared__ float lds_l[NWAVE][16];

    int bg = blockIdx.x;
    int b = bg >> 3, g = bg & 7;
    int tid  = threadIdx.x;
    int w    = tid >> 5;
    int lane = tid & 31;
    int ln   = lane & 15;
    int grp  = lane >> 4;
    int kb   = grp << 1;

    const float* Q = qbuf + ((size_t)(b * NKV + g)) * 16 * HD;
    v2f qf[32];
#pragma unroll
    for (int kk = 0; kk < 32; ++kk)
        qf[kk] = *(const v2f*)(Q + ln * HD + kk * 4 + kb);

    const float scale = 0.08838834764831845f; // 1/sqrt(128)
    v8f acc[8];
#pragma unroll
    for (int i = 0; i < 8; ++i) acc[i] = {};
    float mrun[8], lrun[8];
#pragma unroll
    for (int r = 0; r < 8; ++r) { mrun[r] = -INFINITY; lrun[r] = 0.0f; }

    const float* kbase = kc + ((size_t)b * TT * NKV + g) * HD;
    const float* vbase = vc + ((size_t)b * TT * NKV + g) * HD;
    const size_t tstr = (size_t)NKV * HD; // 1024 floats between key rows

    float* kr = &lds_k[w][0];
    float* vr = &lds_v[w][0];
    unsigned koff = (unsigned)(size_t)kr;
    unsigned voff = (unsigned)(size_t)vr;

    for (int tt = w; tt < NT_TILES; tt += NWAVE) {
        int t0 = tt * 16;
        unsigned rows = (unsigned)(TT - t0) < 16u ? (unsigned)(TT - t0) : 16u;

        // ---- TDM: K tile (16 keys x 128) -> LDS, padded rows ----
        tdm_load_16x128(kr, koff, kbase + (size_t)t0 * tstr, rows);

        // ---- scores S(16x16) = Q @ K^T over HD=128, B-frags from LDS ----
        v8f s = {};
#pragma unroll
        for (int kk = 0; kk < 32; ++kk) {
            v2f bf;
            bf.x = kr[ln * KROW + kk * 4 + kb];
            bf.y = kr[ln * KROW + kk * 4 + kb + 1];
            s = __builtin_amdgcn_wmma_f32_16x16x4_f32(false, qf[kk], false, bf,
                                                      (short)0, s, false, false);
        }
        bool valid = (t0 + ln) < TT;
#pragma unroll
        for (int r = 0; r < 8; ++r) s[r] = valid ? s[r] * scale : -INFINITY;

        // ---- online softmax (row = VGPR index within 16-lane group) ----
        float p[8], corr[8];
#pragma unroll
        for (int r = 0; r < 8; ++r) {
            float rm = s[r];
            rm = fmaxf(rm, __shfl_xor(rm, 1, 32));
            rm = fmaxf(rm, __shfl_xor(rm, 2, 32));
            rm = fmaxf(rm, __shfl_xor(rm, 4, 32));
            rm = fmaxf(rm, __shfl_xor(rm, 8, 32));
            float mn2 = fmaxf(mrun[r], rm);
            float cr  = __expf(mrun[r] - mn2);
            float pv  = __expf(s[r] - mn2);
            float rs  = pv;
            rs += __shfl_xor(rs, 1, 32);
            rs += __shfl_xor(rs, 2, 32);
            rs += __shfl_xor(rs, 4, 32);
            rs += __shfl_xor(rs, 8, 32);
            lrun[r] = lrun[r] * cr + rs;
            mrun[r] = mn2;
            p[r] = pv;
            corr[r] = cr;
        }
#pragma unroll
        for (int n0 = 0; n0 < 8; ++n0)
#pragma unroll
            for (int r = 0; r < 8; ++r) acc[n0][r] *= corr[r];

        // ---- stage P (C-layout) into LDS for A-fragment re-read ----
#pragma unroll
        for (int r = 0; r < 8; ++r) lds_p[w][r + 8 * grp][ln] = p[r];

        // ---- TDM: V tile -> LDS ----
        tdm_load_16x128(vr, voff, vbase + (size_t)t0 * tstr, rows);

        // ---- acc(16x128) += P(16x16) @ V(16x128), B-frags from LDS ----
#pragma unroll
        for (int kk = 0; kk < 4; ++kk) {
            v2f a;
            a.x = lds_p[w][ln][kk * 4 + kb];
            a.y = lds_p[w][ln][kk * 4 + kb + 1];
#pragma unroll
            for (int n0 = 0; n0 < 8; ++n0) {
                v2f bf;
                bf.x = vr[(kk * 4 + kb) * KROW + n0 * 16 + ln];
                bf.y = vr[(kk * 4 + kb + 1) * KROW + n0 * 16 + ln];
                acc[n0] = __builtin_amdgcn_wmma_f32_16x16x4_f32(false, a, false, bf,
                                                                (short)0, acc[n0],
                                                                false, false);
            }
        }
    }

    // ---- dump wave partials ----
#pragma unroll
    for (int n0 = 0; n0 < 8; ++n0)
#pragma unroll
        for (int r = 0; r < 8; ++r)
            lds_acc[w][r + 8 * grp][n0 * 16 + ln] = acc[n0][r];
    if (ln == 0) {
#pragma unroll
        for (int r = 0; r < 8; ++r) {
            lds_m[w][r + 8 * grp] = mrun[r];
            lds_l[w][r + 8 * grp] = lrun[r];
        }
    }
    __syncthreads();

    // ---- combine wave partials, normalize, scatter to attn layout ----
    for (int idx = tid; idx < 16 * HD; idx += NWAVE * 32) {
        int m = idx >> 7, d = idx & 127;
        float M = -INFINITY;
#pragma unroll
        for (int ww = 0; ww < NWAVE; ++ww) M = fmaxf(M, lds_m[ww][m]);
        float L = 0.0f, o = 0.0f;
#pragma unroll
        for (int ww = 0; ww < NWAVE; ++ww) {
            float e = __expf(lds_m[ww][m] - M);
            L += lds_l[ww][m] * e;
            o += lds_acc[ww][m][d] * e;
        }
        o /= L;
        int s_ = m >> 2, r_ = m & 3;
        int h = g * 4 + r_;
        attn[(((size_t)(b * S_ + s_)) * NH + h) * HD + d] = o;
    }
}

// ---------------------------------------------------------------------------
extern "C" void kernel_launch(void* const* d_in, const int* in_sizes, int n_in,
                              void* d_out, int out_size, void* d_ws, size_t ws_size,
                              hipStream_t stream) {
    const int*   positions = (const int*)d_in[0];
    const float* hidden    = (const float*)d_in[1];
    const float* past_k    = (const float*)d_in[2];
    const float* past_v    = (const float*)d_in[3];
    const float* w_qkv     = (const float*)d_in[4];
    const float* w_o       = (const float*)d_in[5];

    float* out  = (float*)d_out;
    float* newk = out + (size_t)B_ * S_ * H_;
    float* newv = newk + (size_t)B_ * TT * NKV * HD;

    float* qkv  = (float*)d_ws;                        // 64*6144
    float* qbuf = qkv + (size_t)B_ * S_ * QKV_N;       // 16*8*16*128
    float* attn = qbuf + (size_t)B_ * NKV * 16 * HD;   // 64*4096

    // 1) stream past KV into new KV head (the ~1.07 GB memory floor)
    copy_past_kernel<<<8192, 256, 0, stream>>>((const float4*)past_k,
                                               (const float4*)past_v,
                                               (float4*)newk, (float4*)newv);
    // 2) QKV projection: (64 x 4096) @ (4096 x 6144)
    dim3 g1(QKV_N / 16, (B_ * S_) / 16);
    gemm_f32_wmma<<<g1, 32, 0, stream>>>(hidden, w_qkv, qkv, QKV_N, H_);
    // 3) RoPE + scatter to qbuf / new_k tail / new_v tail
    int rope_threads = B_ * S_ * NH * (HD / 2) + B_ * S_ * NKV * (HD / 2) + B_ * S_ * NKV * HD;
    rope_scatter_kernel<<<(rope_threads + 255) / 256, 256, 0, stream>>>(
        positions, qkv, qbuf, newk, newv);
    // 4) flash attention per (b, kv head), K/V staged via Tensor Data Mover
    attn_kernel<<<B_ * NKV, NWAVE * 32, 0, stream>>>(qbuf, newk, newv, attn);
    // 5) output projection: (64 x 4096) @ (4096 x 4096)
    dim3 g2(H_ / 16, (B_ * S_) / 16);
    gemm_f32_wmma<<<g2, 32, 0, stream>>>(attn, w_o, out, H_, H_);
}